// XEmbedding_63840393888373
// MI455X (gfx1250) — compile-verified
//
#include <hip/hip_runtime.h>

typedef __attribute__((ext_vector_type(2))) float v2f;
typedef __attribute__((ext_vector_type(4))) float v4f;
typedef __attribute__((ext_vector_type(8))) float v8f;

#define N_NODES   16000
#define N_EDGES   512000
#define EMBED_DIM 28
#define NODE_DIM  128
#define NUM_BASIS 20
#define RSH_DIM   480   // 128*sh0 + 64*3 sh1 + 32*5 sh2

// ---------------------------------------------------------------------------
// Kernel 1: x_scalar = embed_table[at_no] @ W^T + b  via V_WMMA_F32_16X16X4_F32
// Grid: 1000 blocks (16 rows each) x 8 waves (one 16-col N tile per wave).
// K = 28 = 7 x 4 WMMA accumulation steps, all in fp32 (exact path).
// ---------------------------------------------------------------------------
__global__ __launch_bounds__(256) void node_embed_gemm(
    const int*   __restrict__ at_no,        // (16000)
    const float* __restrict__ embed_table,  // (87, 28) row-major
    const float* __restrict__ W,            // (128, 28) row-major == B(K x N) as W[n][k]
    const float* __restrict__ bvec,         // (128)
    float*       __restrict__ x_scalar)     // (16000, 128)
{
    const int wave  = threadIdx.x >> 5;     // N tile 0..7
    const int lane  = threadIdx.x & 31;
    const int half  = lane >> 4;            // 0: lanes 0-15, 1: lanes 16-31
    const int l16   = lane & 15;
    const int m0    = blockIdx.x * 16;      // 16000 = 1000*16 exactly

    // A fragment source: gathered embedding row for M = m0 + l16
    const int   arowi = at_no[m0 + l16];
    const float* arow = embed_table + arowi * EMBED_DIM;
    // B fragment source: B[k][n] = W[nG][k]
    const int    nG   = wave * 16 + l16;
    const float* wrow = W + nG * EMBED_DIM;

    // C init = bias broadcast: every element this lane owns has column nG
    float bias = bvec[nG];
    v8f c;
#pragma unroll
    for (int i = 0; i < 8; ++i) c[i] = bias;

    // A 16x4 f32 layout: VGPR v, half h -> K = v + 2*h ; B 4x16: same K mapping
#pragma unroll
    for (int kk = 0; kk < 7; ++kk) {
        const int kb = kk * 4 + half * 2;
        v2f a; a.x = arow[kb]; a.y = arow[kb + 1];
        v2f b; b.x = wrow[kb]; b.y = wrow[kb + 1];
        c = __builtin_amdgcn_wmma_f32_16x16x4_f32(
                /*neg_a=*/false, a, /*neg_b=*/false, b,
                /*c_mod=*/(short)0, c, /*reuse_a=*/false, /*reuse_b=*/false);
    }

    // C/D 16x16 f32 layout: VGPR r -> M = r + 8*half, N = l16
#pragma unroll
    for (int r = 0; r < 8; ++r) {
        const int mr = m0 + r + half * 8;
        x_scalar[mr * NODE_DIM + nG] = c[r];
    }
}

// ---------------------------------------------------------------------------
// Kernel 2: per-edge rbf / fcut / rsh. One wave per edge; the 983 MB rsh
// tensor is streamed with 16B nontemporal stores (no reuse; bypass L2 rinse).
// ---------------------------------------------------------------------------
__device__ __forceinline__ float pick3(int r, float a0, float a1, float a2) {
    return r == 0 ? a0 : (r == 1 ? a1 : a2);
}
__device__ __forceinline__ float pick5(int r, float a0, float a1, float a2,
                                       float a3, float a4) {
    return r == 0 ? a0 : (r == 1 ? a1 : (r == 2 ? a2 : (r == 3 ? a3 : a4)));
}

__global__ __launch_bounds__(256) void edge_kernel(
    const float* __restrict__ pos,         // (16000, 3)
    const int*   __restrict__ edge_index,  // (2, 512000)
    float*       __restrict__ rbf,         // (512000, 20)
    float*       __restrict__ fcut,        // (512000, 1)
    float*       __restrict__ rsh)         // (512000, 480)
{
    const int e    = blockIdx.x * 8 + (threadIdx.x >> 5);
    const int lane = threadIdx.x & 31;
    if (e >= N_EDGES) return;

    const int s = edge_index[e];
    const int d = edge_index[N_EDGES + e];

    // Reference permutes pos columns [1,2,0] before the diff.
    const float vx = pos[s * 3 + 1] - pos[d * 3 + 1];
    const float vy = pos[s * 3 + 2] - pos[d * 3 + 2];
    const float vz = pos[s * 3 + 0] - pos[d * 3 + 0];

    const float dist = sqrtf(vx * vx + vy * vy + vz * vz);
    const float inv  = 1.0f / dist;
    const float x = vx * inv, y = vy * inv, z = vz * inv;

    const float s3  = 1.7320508075688772f;   // sqrt(3)
    const float s5  = 2.2360679774997896f;   // sqrt(5)
    const float s15 = 3.8729833462074170f;   // sqrt(15)

    const float h10 = s3 * x, h11 = s3 * y, h12 = s3 * z;
    const float h20 = s15 * x * z;
    const float h21 = s15 * x * y;
    const float h22 = s5 * (y * y - 0.5f * (x * x + z * z));
    const float h23 = s15 * y * z;
    const float h24 = 0.5f * s15 * (z * z - x * x);

    // --- rsh row: cols [0,128)=1.0, [128,320)=sh1 tiled, [320,480)=sh2 tiled
    float* rrow = rsh + (size_t)e * RSH_DIM;
#pragma unroll
    for (int i = 0; i < 4; ++i) {
        const int cbase = i * 128 + lane * 4;
        if (cbase < RSH_DIM) {                      // last iter: lanes 0-23 only
            v4f v;
#pragma unroll
            for (int j = 0; j < 4; ++j) {
                const int c = cbase + j;
                float val;
                if (c < 128)      val = 1.0f;
                else if (c < 320) val = pick3((c - 128) % 3, h10, h11, h12);
                else              val = pick5((c - 320) % 5, h20, h21, h22, h23, h24);
                v[j] = val;
            }
            __builtin_nontemporal_store(v, (v4f*)(rrow + cbase));
        }
    }

    // --- rbf: lanes 0..19, n = lane+1
    if (lane < NUM_BASIS) {
        const float n   = (float)(lane + 1);
        const float PIF = 3.14159265358979323846f;
        const float val = 0.63245553203367587f /* sqrt(2/5) */
                          * sinf(n * PIF * dist * 0.2f) * inv;
        __builtin_nontemporal_store(val, rbf + (size_t)e * NUM_BASIS + lane);
    }

    // --- fcut: polynomial cutoff, p=5 -> 1 - 21u^5 + 35u^6 - 15u^7
    if (lane == 0) {
        const float u = dist * 0.2f;
        float f = 0.0f;
        if (u < 1.0f) {
            const float u2 = u * u;
            const float u5 = u2 * u2 * u;
            f = 1.0f + u5 * (-21.0f + u * (35.0f - 15.0f * u));
        }
        fcut[e] = f;
    }
}

// ---------------------------------------------------------------------------
extern "C" void kernel_launch(void* const* d_in, const int* in_sizes, int n_in,
                              void* d_out, int out_size, void* d_ws, size_t ws_size,
                              hipStream_t stream) {
    const int*   at_no       = (const int*)  d_in[0];
    const float* pos         = (const float*)d_in[1];
    const int*   edge_index  = (const int*)  d_in[2];
    const float* embed_table = (const float*)d_in[3];
    const float* W           = (const float*)d_in[4];
    const float* b           = (const float*)d_in[5];

    float* out      = (float*)d_out;
    float* x_scalar = out;                                   // 16000*128
    float* rbf      = out + (size_t)N_NODES * NODE_DIM;      // 512000*20
    float* fcut     = rbf + (size_t)N_EDGES * NUM_BASIS;     // 512000
    float* rsh      = fcut + (size_t)N_EDGES;                // 512000*480

    node_embed_gemm<<<N_NODES / 16, 256, 0, stream>>>(at_no, embed_table, W, b, x_scalar);
    edge_kernel<<<N_EDGES / 8, 256, 0, stream>>>(pos, edge_index, rbf, fcut, rsh);
}